// GAE_17437567222238
// MI455X (gfx1250) — compile-verified
//
#include <hip/hip_runtime.h>

typedef float v2f __attribute__((ext_vector_type(2)));
typedef float v8f __attribute__((ext_vector_type(8)));

#define NNODES 10000
#define FDIM   512
#define H1DIM  32
#define DDIM   16
#define NEDGE  320000
#define NTILE  625            // 10000 / 16

// ---------------------------------------------------------------------------
// zero-init a float range
__global__ void k_zero(float* __restrict__ p, int n) {
    int i = blockIdx.x * blockDim.x + threadIdx.x;
    if (i < n) p[i] = 0.0f;
}

// ---------------------------------------------------------------------------
// xw1[10000,32] = x[10000,512] @ W1[512,32]
// one wave computes a 16x32 output slab via V_WMMA_F32_16X16X4_F32
__global__ void k_gemm1(const float* __restrict__ x, const float* __restrict__ W1,
                        float* __restrict__ xw1) {
    int wave = (blockIdx.x * blockDim.x + threadIdx.x) >> 5;
    int lane = threadIdx.x & 31;
    if (wave >= NTILE) return;               // whole-wave guard: EXEC stays all-1s
    int r0   = wave << 4;
    int half = lane >> 4;                    // 0: K even pair, 1: K odd pair (+2)
    int l15  = lane & 15;
    const float* arow = x + (size_t)(r0 + l15) * FDIM;
    v8f c0 = {}; v8f c1 = {};
    for (int k0 = 0; k0 < FDIM; k0 += 4) {
        int kb = k0 + half * 2;
        // x is 20 MB read exactly once -> stream it (TH_NT), keep W1 cached (RT)
        v2f a = __builtin_nontemporal_load((const v2f*)(arow + kb)); // A[m, kb..kb+1]
        v2f b0, b1;
        b0.x = W1[(size_t)kb       * H1DIM + l15];        // B[kb,   n]
        b0.y = W1[(size_t)(kb + 1) * H1DIM + l15];        // B[kb+1, n]
        b1.x = W1[(size_t)kb       * H1DIM + 16 + l15];
        b1.y = W1[(size_t)(kb + 1) * H1DIM + 16 + l15];
        c0 = __builtin_amdgcn_wmma_f32_16x16x4_f32(false, a, false, b0, (short)0, c0, false, false);
        c1 = __builtin_amdgcn_wmma_f32_16x16x4_f32(false, a, false, b1, (short)0, c1, false, false);
    }
#pragma unroll
    for (int v = 0; v < 8; ++v) {
        int row = r0 + v + half * 8;
        xw1[(size_t)row * H1DIM + l15]      = c0[v];
        xw1[(size_t)row * H1DIM + 16 + l15] = c1[v];
    }
}

// ---------------------------------------------------------------------------
// h1[r,:] += v * xw1[c,:]   (COO scatter; wave per edge, lane = feature 0..31)
__global__ void k_spmm1(const int* __restrict__ rows, const int* __restrict__ cols,
                        const float* __restrict__ vals,
                        const float* __restrict__ xw1, float* __restrict__ h1) {
    int e    = (blockIdx.x * blockDim.x + threadIdx.x) >> 5;
    int lane = threadIdx.x & 31;
    if (e >= NEDGE) return;
    int   r = rows[e];
    int   c = cols[e];
    float v = vals[e];
    atomicAdd(&h1[(size_t)r * H1DIM + lane], v * xw1[(size_t)c * H1DIM + lane]);
}

// ---------------------------------------------------------------------------
// h1w2[10000,16] = relu(h1)[10000,32] @ W2[32,16]   (relu fused into A load)
__global__ void k_gemm2(const float* __restrict__ h1, const float* __restrict__ W2,
                        float* __restrict__ h1w2) {
    int wave = (blockIdx.x * blockDim.x + threadIdx.x) >> 5;
    int lane = threadIdx.x & 31;
    if (wave >= NTILE) return;
    int r0   = wave << 4;
    int half = lane >> 4;
    int l15  = lane & 15;
    const float* arow = h1 + (size_t)(r0 + l15) * H1DIM;
    v8f c = {};
#pragma unroll
    for (int k0 = 0; k0 < H1DIM; k0 += 4) {
        int kb = k0 + half * 2;
        v2f a = *(const v2f*)(arow + kb);
        a.x = fmaxf(a.x, 0.0f);
        a.y = fmaxf(a.y, 0.0f);
        v2f b;
        b.x = W2[(size_t)kb       * DDIM + l15];
        b.y = W2[(size_t)(kb + 1) * DDIM + l15];
        c = __builtin_amdgcn_wmma_f32_16x16x4_f32(false, a, false, b, (short)0, c, false, false);
    }
#pragma unroll
    for (int v = 0; v < 8; ++v) {
        int row = r0 + v + half * 8;
        h1w2[(size_t)row * DDIM + l15] = c[v];
    }
}

// ---------------------------------------------------------------------------
// z[r,:] += v * h1w2[c,:]   (thread per (edge, feature), D=16)
__global__ void k_spmm2(const int* __restrict__ rows, const int* __restrict__ cols,
                        const float* __restrict__ vals,
                        const float* __restrict__ h1w2, float* __restrict__ z) {
    int t = blockIdx.x * blockDim.x + threadIdx.x;
    int e = t >> 4;
    int f = t & 15;
    if (e >= NEDGE) return;
    int   r = rows[e];
    int   c = cols[e];
    float v = vals[e];
    atomicAdd(&z[(size_t)r * DDIM + f], v * h1w2[(size_t)c * DDIM + f]);
}

// ---------------------------------------------------------------------------
// relu z in place (for decoder) + emit z as output #1
__global__ void k_relu_z(float* __restrict__ z, float* __restrict__ out_z) {
    int i = blockIdx.x * blockDim.x + threadIdx.x;
    if (i < NNODES * DDIM) {
        float v = fmaxf(z[i], 0.0f);
        z[i] = v;
        out_z[i] = v;
    }
}

// ---------------------------------------------------------------------------
// a_hat = sigmoid(z @ z^T), 16x16 tile per wave, K=16 -> 4 WMMA f32 ops.
// 400 MB streamed output (2x L2 capacity): use non-temporal stores so the
// write stream doesn't evict the hot 640 KB z matrix from L2/WGP$.
__global__ void k_decoder(const float* __restrict__ z, float* __restrict__ ahat) {
    long wave = ((long)blockIdx.x * blockDim.x + threadIdx.x) >> 5;
    int  lane = threadIdx.x & 31;
    if (wave >= (long)NTILE * NTILE) return;
    int tr = (int)(wave / NTILE);
    int tc = (int)(wave % NTILE);
    int r0 = tr << 4, c0 = tc << 4;
    int half = lane >> 4;
    int l15  = lane & 15;
    const float* zr = z + (size_t)(r0 + l15) * DDIM;   // A rows
    const float* zc = z + (size_t)(c0 + l15) * DDIM;   // B cols (B[k,n] = z[n,k])
    v8f acc = {};
#pragma unroll
    for (int k0 = 0; k0 < DDIM; k0 += 4) {
        v2f a = *(const v2f*)(zr + k0 + half * 2);
        v2f b = *(const v2f*)(zc + k0 + half * 2);
        acc = __builtin_amdgcn_wmma_f32_16x16x4_f32(false, a, false, b, (short)0, acc, false, false);
    }
#pragma unroll
    for (int v = 0; v < 8; ++v) {
        int row = r0 + v + half * 8;
        float s = 1.0f / (1.0f + __expf(-acc[v]));
        __builtin_nontemporal_store(s, &ahat[(size_t)row * NNODES + c0 + l15]);
    }
}

// ---------------------------------------------------------------------------
extern "C" void kernel_launch(void* const* d_in, const int* in_sizes, int n_in,
                              void* d_out, int out_size, void* d_ws, size_t ws_size,
                              hipStream_t stream) {
    const float* x        = (const float*)d_in[0];
    const float* adj_vals = (const float*)d_in[1];
    const float* W1       = (const float*)d_in[2];
    const float* W2       = (const float*)d_in[3];
    const int*   rows     = (const int*)d_in[4];   // harness integer convention
    const int*   cols     = (const int*)d_in[5];

    float* out_z    = (float*)d_out;               // [10000,16]
    float* out_ahat = out_z + (size_t)NNODES * DDIM;

    float* ws   = (float*)d_ws;
    float* xw1  = ws;                       // 320000 floats
    float* h1   = ws + 320000;              // 320000 floats (atomic accum)
    float* h1w2 = ws + 640000;              // 160000 floats
    float* zacc = ws + 800000;              // 160000 floats (atomic accum)

    // zero the two atomic accumulators (and h1w2, harmless): 640000 floats
    k_zero<<<(640000 + 255) / 256, 256, 0, stream>>>(ws + 320000, 640000);

    // stage 1: dense GEMM x@W1 (625 waves)
    k_gemm1<<<(NTILE * 32 + 255) / 256, 256, 0, stream>>>(x, W1, xw1);

    // stage 2: SpMM scatter into h1 (wave per edge)
    k_spmm1<<<(NEDGE * 32) / 256, 256, 0, stream>>>(rows, cols, adj_vals, xw1, h1);

    // stage 3: relu(h1)@W2 (625 waves)
    k_gemm2<<<(NTILE * 32 + 255) / 256, 256, 0, stream>>>(h1, W2, h1w2);

    // stage 4: SpMM scatter into zacc
    k_spmm2<<<(NEDGE * 16) / 256, 256, 0, stream>>>(rows, cols, adj_vals, h1w2, zacc);

    // stage 5: relu z, emit output #1
    k_relu_z<<<(NNODES * DDIM + 255) / 256, 256, 0, stream>>>(zacc, out_z);

    // stage 6: decoder sigmoid(z z^T) -> 400 MB output, store-BW bound
    long ntiles = (long)NTILE * NTILE;                 // 390625 waves
    int  nblk   = (int)((ntiles + 7) / 8);             // 8 waves / 256-thread block
    k_decoder<<<nblk, 256, 0, stream>>>(zacc, out_ahat);
}